// RayBoxIntersection_83837761618478
// MI455X (gfx1250) — compile-verified
//
#include <hip/hip_runtime.h>

typedef float __attribute__((ext_vector_type(4))) f32x4;

// Per-ray slab test, matching the reference's operation order:
// inv_d = 1/d; t_min=(bmin-o)*inv_d; t_max=(bmax-o)*inv_d;
// t_near = max(min(t_min,t_max)); t_far = min(max(t_min,t_max));
// hit = (t_far > t_near) && (t_far > 0)
__device__ __forceinline__ void intersect1(
    float ox, float oy, float oz,
    float dx, float dy, float dz,
    float bminx, float bminy, float bminz,
    float bmaxx, float bmaxy, float bmaxz,
    float& z_in, float& z_out, float& hitf,
    float& pix, float& piy, float& piz,
    float& pox, float& poy, float& poz)
{
    const float ix = 1.0f / dx;   // IEEE divide: +-inf for axis-parallel rays
    const float iy = 1.0f / dy;
    const float iz = 1.0f / dz;

    const float tnx = (bminx - ox) * ix;
    const float tny = (bminy - oy) * iy;
    const float tnz = (bminz - oz) * iz;
    const float txx = (bmaxx - ox) * ix;
    const float txy = (bmaxy - oy) * iy;
    const float txz = (bmaxz - oz) * iz;

    const float t0x = fminf(tnx, txx), t1x = fmaxf(tnx, txx);
    const float t0y = fminf(tny, txy), t1y = fmaxf(tny, txy);
    const float t0z = fminf(tnz, txz), t1z = fmaxf(tnz, txz);

    const float tnear = fmaxf(t0x, fmaxf(t0y, t0z));  // v_max3_num_f32
    const float tfar  = fminf(t1x, fminf(t1y, t1z));  // v_min3_num_f32

    const bool hit = (tfar > tnear) && (tfar > 0.0f);

    z_in  = hit ? tnear : 0.0f;
    z_out = hit ? tfar  : 0.0f;
    hitf  = hit ? 1.0f  : 0.0f;

    pix = hit ? (ox + dx * z_in)  : 0.0f;
    piy = hit ? (oy + dy * z_in)  : 0.0f;
    piz = hit ? (oz + dz * z_in)  : 0.0f;
    pox = hit ? (ox + dx * z_out) : 0.0f;
    poy = hit ? (oy + dy * z_out) : 0.0f;
    poz = hit ? (oz + dz * z_out) : 0.0f;
}

__global__ __launch_bounds__(256) void RayBoxIntersection_83837761618478_kernel(
    const float* __restrict__ org,
    const float* __restrict__ dir,
    const float* __restrict__ box,
    float* __restrict__ out,
    const long long nRays)
{
    const long long tid = (long long)blockIdx.x * 256 + threadIdx.x;
    const long long g   = tid * 4;                 // first ray of this lane's group
    if (g >= nRays) return;

    // Uniform box bounds -> scalar loads (s_load), broadcast to all lanes.
    const float bminx = box[0], bminy = box[1], bminz = box[2];
    const float bmaxx = box[3], bmaxy = box[4], bmaxz = box[5];

    // Output tuple flattened in return order.
    float* z_in_out  = out;
    float* z_out_out = out + nRays;
    float* pin_out   = out + 2 * nRays;
    float* pout_out  = out + 5 * nRays;
    float* hit_out   = out + 8 * nRays;

    if (g + 4 <= nRays) {
        // ---- vector path: 4 rays, all traffic as 16B-aligned b128 NT ----
        const f32x4* op = (const f32x4*)(org + g * 3);   // byte offset = tid*48 (16B aligned)
        const f32x4* dp = (const f32x4*)(dir + g * 3);

        float o[12], d[12];
        ((f32x4*)o)[0] = __builtin_nontemporal_load(op + 0);
        ((f32x4*)o)[1] = __builtin_nontemporal_load(op + 1);
        ((f32x4*)o)[2] = __builtin_nontemporal_load(op + 2);
        ((f32x4*)d)[0] = __builtin_nontemporal_load(dp + 0);
        ((f32x4*)d)[1] = __builtin_nontemporal_load(dp + 1);
        ((f32x4*)d)[2] = __builtin_nontemporal_load(dp + 2);

        float zi[4], zo[4], hm[4];
        float pi[12], po[12];

        #pragma unroll
        for (int r = 0; r < 4; ++r) {
            intersect1(o[r * 3 + 0], o[r * 3 + 1], o[r * 3 + 2],
                       d[r * 3 + 0], d[r * 3 + 1], d[r * 3 + 2],
                       bminx, bminy, bminz, bmaxx, bmaxy, bmaxz,
                       zi[r], zo[r], hm[r],
                       pi[r * 3 + 0], pi[r * 3 + 1], pi[r * 3 + 2],
                       po[r * 3 + 0], po[r * 3 + 1], po[r * 3 + 2]);
        }

        __builtin_nontemporal_store(((f32x4*)zi)[0], (f32x4*)(z_in_out  + g));
        __builtin_nontemporal_store(((f32x4*)zo)[0], (f32x4*)(z_out_out + g));
        __builtin_nontemporal_store(((f32x4*)hm)[0], (f32x4*)(hit_out   + g));

        f32x4* pip = (f32x4*)(pin_out  + g * 3);
        f32x4* pop = (f32x4*)(pout_out + g * 3);
        __builtin_nontemporal_store(((f32x4*)pi)[0], pip + 0);
        __builtin_nontemporal_store(((f32x4*)pi)[1], pip + 1);
        __builtin_nontemporal_store(((f32x4*)pi)[2], pip + 2);
        __builtin_nontemporal_store(((f32x4*)po)[0], pop + 0);
        __builtin_nontemporal_store(((f32x4*)po)[1], pop + 1);
        __builtin_nontemporal_store(((f32x4*)po)[2], pop + 2);
    } else {
        // ---- scalar tail (unused for B*R multiple of 4, kept for safety) ----
        for (long long r = g; r < nRays; ++r) {
            float zi, zo, hf, pix, piy, piz, pox, poy, poz;
            intersect1(org[r * 3 + 0], org[r * 3 + 1], org[r * 3 + 2],
                       dir[r * 3 + 0], dir[r * 3 + 1], dir[r * 3 + 2],
                       bminx, bminy, bminz, bmaxx, bmaxy, bmaxz,
                       zi, zo, hf, pix, piy, piz, pox, poy, poz);
            z_in_out[r]  = zi;
            z_out_out[r] = zo;
            hit_out[r]   = hf;
            pin_out[r * 3 + 0] = pix; pin_out[r * 3 + 1] = piy; pin_out[r * 3 + 2] = piz;
            pout_out[r * 3 + 0] = pox; pout_out[r * 3 + 1] = poy; pout_out[r * 3 + 2] = poz;
        }
    }
}

extern "C" void kernel_launch(void* const* d_in, const int* in_sizes, int n_in,
                              void* d_out, int out_size, void* d_ws, size_t ws_size,
                              hipStream_t stream) {
    const float* origins    = (const float*)d_in[0];
    const float* directions = (const float*)d_in[1];
    const float* box        = (const float*)d_in[2];
    float* out = (float*)d_out;

    const long long nRays    = (long long)in_sizes[0] / 3;   // B*R = 4,194,304
    const long long nThreads = (nRays + 3) / 4;
    const int block = 256;                                   // 8 wave32s per WGP pass
    const long long grid = (nThreads + block - 1) / block;   // 4096 blocks

    RayBoxIntersection_83837761618478_kernel<<<(unsigned)grid, block, 0, stream>>>(
        origins, directions, box, out, nRays);
}